// LookAheadMask_1314259993026
// MI455X (gfx1250) — compile-verified
//
#include <hip/hip_runtime.h>

// out[b, r, c] = (c > r) ? 1.0f : x[b, r, c]   for x of shape (B, S, S), fp32.
//
// Memory-bound kernel: ~512 MiB stores + ~256 MiB loads (upper triangle is
// never read). float4 (b128) accesses, non-temporal hints for the streaming
// traffic (working set >> 192 MB L2).

typedef float float4_t __attribute__((ext_vector_type(4)));

__global__ __launch_bounds__(256) void lookahead_mask_kernel(
    const float4_t* __restrict__ x,
    float4_t* __restrict__ out,
    unsigned int S)  // row length (4096); vec-per-row = S/4
{
    const unsigned int vpr = S >> 2;                       // float4s per row
    const unsigned int j   = blockIdx.x * blockDim.x + threadIdx.x; // vec col
    const unsigned int r   = blockIdx.y;                   // row within matrix
    const unsigned int b   = blockIdx.z;                   // batch

    if (j >= vpr) return;

    const size_t idx = ((size_t)b * S + r) * (size_t)vpr + j;
    const unsigned int c0 = j << 2;                        // first column of this vec

    float4_t v;
    if (c0 > r) {
        // Entire float4 strictly above the diagonal: write 1.0, skip the load.
        v = (float4_t){1.0f, 1.0f, 1.0f, 1.0f};
    } else {
        float4_t in = __builtin_nontemporal_load(&x[idx]); // th:TH_NT streaming read
        if (c0 + 3u <= r) {
            // Entirely on/below the diagonal: straight copy.
            v = in;
        } else {
            // Straddles the diagonal (at most one vec per row): per-lane blend.
            v.x = in.x;                                    // c0 <= r guaranteed here
            v.y = (c0 + 1u > r) ? 1.0f : in.y;
            v.z = (c0 + 2u > r) ? 1.0f : in.z;
            v.w = (c0 + 3u > r) ? 1.0f : in.w;
        }
    }
    __builtin_nontemporal_store(v, &out[idx]);             // th:TH_NT streaming write
}

extern "C" void kernel_launch(void* const* d_in, const int* in_sizes, int n_in,
                              void* d_out, int out_size, void* d_ws, size_t ws_size,
                              hipStream_t stream) {
    (void)n_in; (void)d_ws; (void)ws_size; (void)out_size;

    const unsigned int S = 4096;                 // reference: S = 4096
    const size_t total   = (size_t)in_sizes[0];  // B * S * S
    const unsigned int B = (unsigned int)(total / ((size_t)S * S)); // 8

    const unsigned int vpr    = S / 4;           // 1024 float4 per row
    const unsigned int block  = 256;
    dim3 grid((vpr + block - 1) / block, S, B);  // (4, 4096, 8)

    lookahead_mask_kernel<<<grid, dim3(block, 1, 1), 0, stream>>>(
        (const float4_t*)d_in[0], (float4_t*)d_out, S);
}